// GSA_33423435497605
// MI455X (gfx1250) — compile-verified
//
#include <hip/hip_runtime.h>

// ---------------------------------------------------------------------------
// Linear attention on CDNA5, fp32 WMMA 16x16x4 (memory-bound: precision free).
// Pipeline:
//   k1:  ctx_unnorm[b,h,64,64] = exp(Wk_h X) (Wv_h X)^T  ; rowsums S (atomics)
//   kf1: Wfused[b, o, h*64+d] = sum_e Wout[o,he] * ctx[d,e]/S[d]
//   kf2: Wbig[b, 256,256]     = Wfused_b @ Wq          (readout fully folded)
//   k3:  out = Wbig_b @ img + b_out                    (img read exactly once)
// Staging uses gfx1250 async global->LDS copies, double buffered.
// ---------------------------------------------------------------------------

typedef __attribute__((ext_vector_type(2))) float v2f;
typedef __attribute__((ext_vector_type(8))) float v8f;

#define HEADS 8
#define DK    64
#define C_IN  256
#define NPIX  16384
#define BATCH 8
#define DHID  512
#define KC    32

// ---- WMMA f32 16x16x4: A/B frag (row=l16, k=2*half+j); C frag (m=8*half+j, n=l16)
__device__ __forceinline__ v8f wmma4(v2f a, v2f b, v8f c) {
#if __has_builtin(__builtin_amdgcn_wmma_f32_16x16x4_f32)
  return __builtin_amdgcn_wmma_f32_16x16x4_f32(
      false, a, false, b, (short)0, c, false, false);
#else
  c[0] += a[0] * b[0];
  c[1] += a[1] * b[1];
  return c;
#endif
}

// ---- gfx1250 async global->LDS (ASYNCcnt) with sync fallback --------------
#if __has_builtin(__builtin_amdgcn_global_load_async_to_lds_b32)
#define ASYNC_LDS 1
#endif

typedef __attribute__((address_space(1))) int as1_int;
typedef __attribute__((address_space(3))) int as3_int;

__device__ __forceinline__ void async_g2l_b32(const float* g, float* l) {
#if defined(ASYNC_LDS)
  __builtin_amdgcn_global_load_async_to_lds_b32(
      (as1_int*)g, (as3_int*)l, 0, 0);
#else
  *l = *g;
#endif
}

__device__ __forceinline__ void async_wait0() {
#if defined(ASYNC_LDS)
#if __has_builtin(__builtin_amdgcn_s_wait_asynccnt)
  __builtin_amdgcn_s_wait_asynccnt(0);
#else
  asm volatile("s_wait_asynccnt 0x0" ::: "memory");
#endif
#endif
}

// Stage KC x 128 chunk of X (row-major [C_IN][NPIX]) transposed into
// XT[token][channel] via per-lane async scatter (16 copies/lane = chunk/wave).
__device__ __forceinline__ void issue_stage_xt(float (*XTb)[KC + 2],
                                               const float* __restrict__ X,
                                               int kc, int t0, int wv, int lane) {
  #pragma unroll
  for (int ci = 0; ci < 4; ++ci) {
    const int c = wv * 4 + ci;
    const float* g = X + (size_t)(kc + c) * NPIX + t0 + lane;
    #pragma unroll
    for (int tb = 0; tb < 4; ++tb)
      async_g2l_b32(g + tb * 32, &XTb[tb * 32 + lane][c]);
  }
}

// Synchronous transpose staging for small weight matrices (row stride ld).
__device__ __forceinline__ void stage_T(float (*T)[KC + 2],
                                        const float* __restrict__ src,
                                        int ld, int kc, int c0) {
  #pragma unroll
  for (int r = 0; r < 4; ++r) {
    int idx = threadIdx.x + r * 256;
    int row = idx >> 5;
    int c4 = (idx & 31) << 2;
    float4 vd = *(const float4*)(src + (size_t)(kc + row) * ld + c0 + c4);
    T[c4 + 0][row] = vd.x; T[c4 + 1][row] = vd.y;
    T[c4 + 2][row] = vd.z; T[c4 + 3][row] = vd.w;
  }
}

// ---------------------------------------------------------------------------
// Kernel 1: per (b,h, 1024-token slab): ctx partial + rowsum partial.
// W fragments hoisted to registers (loaded once); X async double-buffered.
// ---------------------------------------------------------------------------
#define K1_TS  128
#define K1_SUB 8

__global__ __launch_bounds__(256) void k1_context(
    const float* __restrict__ img, const float* __restrict__ w_qkv,
    float* __restrict__ S, float* __restrict__ CTX) {
  __shared__ float XT[2][K1_TS][KC + 2];
  __shared__ float KV[2][DK][K1_TS + 4];   // [0]=exp(K), [1]=V
  __shared__ float RS[DK][4];

  const int bh = blockIdx.x, b = bh >> 3, h = bh & 7;
  const int tbase = blockIdx.y * (K1_TS * K1_SUB);
  const int lane = threadIdx.x & 31, wv = threadIdx.x >> 5;
  const int l16 = lane & 15, half = lane >> 4;

  const float* X  = img + (size_t)b * C_IN * NPIX;
  const float* Wk = w_qkv + (size_t)(DHID + h * DK) * C_IN;
  const float* Wv = w_qkv + (size_t)(2 * DHID + h * DK) * C_IN;

  const int m0 = wv * 16;  // wave's m-tile of stacked [K;V]
  const float* Arow = (m0 < DK) ? (Wk + (size_t)(m0 + l16) * C_IN)
                                : (Wv + (size_t)(m0 - DK + l16) * C_IN);

  // kick off first async chunk, overlap with hoisting W into registers
  issue_stage_xt(XT[0], X, 0, tbase, wv, lane);
  v2f areg[64];
  #pragma unroll
  for (int i = 0; i < 64; ++i)
    areg[i] = *(const v2f*)(Arow + i * 4 + 2 * half);

  const int g2_m0 = (wv & 3) * 16;
  const int g2_n0 = (wv >> 2) * 32;
  v8f ctx_acc[2] = {};
  float rsum = 0.f;

  for (int st = 0; st < K1_SUB; ++st) {
    v8f acc[8] = {};
    #pragma unroll
    for (int ck = 0; ck < 8; ++ck) {
      const int f = st * 8 + ck;
      async_wait0();
      __syncthreads();
      if (f + 1 < K1_SUB * 8) {
        const int fn = f + 1;
        issue_stage_xt(XT[fn & 1], X, (fn & 7) * KC,
                       tbase + (fn >> 3) * K1_TS, wv, lane);
      }
      const float (*XB)[KC + 2] = XT[f & 1];
      #pragma unroll
      for (int ks = 0; ks < KC; ks += 4) {
        v2f a = areg[ck * 8 + (ks >> 2)];
        #pragma unroll
        for (int nt = 0; nt < 8; ++nt) {
          v2f bf = *(const v2f*)&XB[nt * 16 + l16][ks + 2 * half];
          acc[nt] = wmma4(a, bf, acc[nt]);
        }
      }
    }
    __syncthreads();
    {  // spill: exp() on K rows, raw V rows
      const int kvsel = (m0 < DK) ? 0 : 1;
      const int mrow = (m0 < DK) ? m0 : m0 - DK;
      #pragma unroll
      for (int nt = 0; nt < 8; ++nt)
        #pragma unroll
        for (int j = 0; j < 8; ++j) {
          float v = acc[nt][j];
          if (kvsel == 0) v = __expf(v);
          KV[kvsel][mrow + 8 * half + j][nt * 16 + l16] = v;
        }
    }
    __syncthreads();
    {  // rowsum partials
      int row = threadIdx.x >> 2, q = threadIdx.x & 3;
      float s = 0.f;
      #pragma unroll
      for (int i = 0; i < 32; ++i) s += KV[0][row][q * 32 + i];
      rsum += s;
    }
    // ctx += expK(64x128) @ V^T(128x64)
    #pragma unroll 4
    for (int ks = 0; ks < K1_TS; ks += 4) {
      v2f a = *(const v2f*)&KV[0][g2_m0 + l16][ks + 2 * half];
      #pragma unroll
      for (int i = 0; i < 2; ++i) {
        v2f bf = *(const v2f*)&KV[1][g2_n0 + i * 16 + l16][ks + 2 * half];
        ctx_acc[i] = wmma4(a, bf, ctx_acc[i]);
      }
    }
    __syncthreads();
  }

  RS[threadIdx.x >> 2][threadIdx.x & 3] = rsum;
  __syncthreads();
  if (threadIdx.x < DK) {
    float s = RS[threadIdx.x][0] + RS[threadIdx.x][1] +
              RS[threadIdx.x][2] + RS[threadIdx.x][3];
    atomicAdd(&S[bh * DK + threadIdx.x], s);
  }
  float* ctxp = CTX + (size_t)bh * DK * DK;
  #pragma unroll
  for (int i = 0; i < 2; ++i)
    #pragma unroll
    for (int j = 0; j < 8; ++j)
      atomicAdd(&ctxp[(g2_m0 + 8 * half + j) * DK + g2_n0 + i * 16 + l16],
                ctx_acc[i][j]);
}

// ---------------------------------------------------------------------------
// kf1: Wfused_b[o, h*64+d] = sum_e Wout[o,h*64+e] * (ctx[d,e]/S[d])
// (normalization folded into the LDS staging load)
// ---------------------------------------------------------------------------
__global__ __launch_bounds__(256) void kf1_fold(
    const float* __restrict__ w_out, const float* __restrict__ CTX,
    const float* __restrict__ S, float* __restrict__ WF) {
  __shared__ float CX[DK][68];   // CX[d][e] = ctx[d][e]/S[d]
  const int bh = blockIdx.x, b = bh >> 3, h = bh & 7;
  const int lane = threadIdx.x & 31, wv = threadIdx.x >> 5;
  const int l16 = lane & 15, half = lane >> 4;

  {
    const float* cp = CTX + (size_t)bh * DK * DK;
    const float* sp = S + bh * DK;
    #pragma unroll
    for (int r = 0; r < 16; ++r) {
      int idx = threadIdx.x + r * 256;   // = d*64 + e
      int d = idx >> 6;
      CX[d][idx & 63] = cp[idx] / sp[d];
    }
  }
  __syncthreads();

  v8f facc[2][4] = {};
  #pragma unroll
  for (int ks = 0; ks < DK; ks += 4) {
    v2f a0 = *(const v2f*)(w_out + (size_t)(wv * 32 + l16) * DHID +
                           h * DK + ks + 2 * half);
    v2f a1 = *(const v2f*)(w_out + (size_t)(wv * 32 + 16 + l16) * DHID +
                           h * DK + ks + 2 * half);
    #pragma unroll
    for (int nt = 0; nt < 4; ++nt) {
      v2f bf = *(const v2f*)&CX[nt * 16 + l16][ks + 2 * half];   // B = ctx^T
      facc[0][nt] = wmma4(a0, bf, facc[0][nt]);
      facc[1][nt] = wmma4(a1, bf, facc[1][nt]);
    }
  }
  float* wf = WF + (size_t)b * C_IN * DHID;   // [256][512]
  #pragma unroll
  for (int i = 0; i < 2; ++i)
    #pragma unroll
    for (int nt = 0; nt < 4; ++nt)
      #pragma unroll
      for (int j = 0; j < 8; ++j)
        wf[(size_t)(wv * 32 + i * 16 + 8 * half + j) * DHID + h * DK +
           nt * 16 + l16] = facc[i][nt][j];
}

// ---------------------------------------------------------------------------
// kf2: Wbig_b[256 x 128-slice] = Wfused_b[256x512] @ Wq[512x256]
// ---------------------------------------------------------------------------
__global__ __launch_bounds__(256) void kf2_wbig(
    const float* __restrict__ WF, const float* __restrict__ w_qkv,
    float* __restrict__ WBIG) {
  __shared__ float QT[128][KC + 2];
  const int b = blockIdx.x;
  const int c0 = blockIdx.y * 128;
  const int lane = threadIdx.x & 31, wv = threadIdx.x >> 5;
  const int l16 = lane & 15, half = lane >> 4;
  const float* wf = WF + (size_t)b * C_IN * DHID;

  v8f acc[2][8] = {};
  for (int kc = 0; kc < DHID; kc += KC) {
    __syncthreads();
    stage_T(QT, w_qkv, C_IN, kc, c0);   // Wq = first 512 rows of w_qkv
    __syncthreads();
    #pragma unroll
    for (int ks = 0; ks < KC; ks += 4) {
      v2f a0 = *(const v2f*)(wf + (size_t)(wv * 32 + l16) * DHID +
                             kc + ks + 2 * half);
      v2f a1 = *(const v2f*)(wf + (size_t)(wv * 32 + 16 + l16) * DHID +
                             kc + ks + 2 * half);
      #pragma unroll
      for (int nt = 0; nt < 8; ++nt) {
        v2f bf = *(const v2f*)&QT[nt * 16 + l16][ks + 2 * half];
        acc[0][nt] = wmma4(a0, bf, acc[0][nt]);
        acc[1][nt] = wmma4(a1, bf, acc[1][nt]);
      }
    }
  }
  float* wb = WBIG + (size_t)b * C_IN * C_IN;
  #pragma unroll
  for (int i = 0; i < 2; ++i)
    #pragma unroll
    for (int nt = 0; nt < 8; ++nt)
      #pragma unroll
      for (int j = 0; j < 8; ++j)
        wb[(size_t)(wv * 32 + i * 16 + 8 * half + j) * C_IN + c0 +
           nt * 16 + l16] = acc[i][nt][j];
}

// ---------------------------------------------------------------------------
// k3: out_b = Wbig_b @ img_b + b_out.  Wbig resident in LDS (266KB), X chunks
// async double-buffered.  img is read exactly once from HBM.
// ---------------------------------------------------------------------------
#define K3_TB  512
#define K3_SUB 4

__global__ __launch_bounds__(256) void k3_out(
    const float* __restrict__ img, const float* __restrict__ WBIG,
    const float* __restrict__ b_out, float* __restrict__ out) {
  __shared__ float WB[C_IN][260];
  __shared__ float XT[2][128][KC + 2];
  const int b = blockIdx.x;
  const int tbase = blockIdx.y * K3_TB;
  const int lane = threadIdx.x & 31, wv = threadIdx.x >> 5;
  const int l16 = lane & 15, half = lane >> 4;
  const float* X = img + (size_t)b * C_IN * NPIX;

  issue_stage_xt(XT[0], X, 0, tbase, wv, lane);
  {  // one-time Wbig -> LDS
    const float* wb = WBIG + (size_t)b * C_IN * C_IN;
    #pragma unroll
    for (int r = 0; r < 64; ++r) {
      int idx = threadIdx.x + r * 256;   // float4 slot
      int o = idx >> 6;
      int c4 = (idx & 63) << 2;
      float4 vd = *(const float4*)(wb + (size_t)o * C_IN + c4);
      WB[o][c4 + 0] = vd.x; WB[o][c4 + 1] = vd.y;
      WB[o][c4 + 2] = vd.z; WB[o][c4 + 3] = vd.w;
    }
  }

  for (int st = 0; st < K3_SUB; ++st) {
    const int t0 = tbase + st * 128;
    v8f acc[2][8] = {};
    for (int ck = 0; ck < 8; ++ck) {
      const int f = st * 8 + ck;
      async_wait0();
      __syncthreads();
      if (f + 1 < K3_SUB * 8) {
        const int fn = f + 1;
        issue_stage_xt(XT[fn & 1], X, (fn & 7) * KC,
                       tbase + (fn >> 3) * 128, wv, lane);
      }
      const float (*XB)[KC + 2] = XT[f & 1];
      const int kc = ck * KC;
      #pragma unroll
      for (int ks = 0; ks < KC; ks += 4) {
        v2f a0 = *(const v2f*)&WB[wv * 32 + l16][kc + ks + 2 * half];
        v2f a1 = *(const v2f*)&WB[wv * 32 + 16 + l16][kc + ks + 2 * half];
        #pragma unroll
        for (int nt = 0; nt < 8; ++nt) {
          v2f bf = *(const v2f*)&XB[nt * 16 + l16][ks + 2 * half];
          acc[0][nt] = wmma4(a0, bf, acc[0][nt]);
          acc[1][nt] = wmma4(a1, bf, acc[1][nt]);
        }
      }
    }
    #pragma unroll
    for (int i = 0; i < 2; ++i)
      #pragma unroll
      for (int nt = 0; nt < 8; ++nt)
        #pragma unroll
        for (int j = 0; j < 8; ++j) {
          int m = wv * 32 + i * 16 + 8 * half + j;
          int n = nt * 16 + l16;
          out[(size_t)(b * C_IN + m) * NPIX + t0 + n] =
              acc[i][nt][j] + b_out[m];
        }
  }
}

// ---------------------------------------------------------------------------
extern "C" void kernel_launch(void* const* d_in, const int* in_sizes, int n_in,
                              void* d_out, int out_size, void* d_ws,
                              size_t ws_size, hipStream_t stream) {
  (void)in_sizes; (void)n_in; (void)out_size; (void)ws_size;
  const float* img   = (const float*)d_in[0];
  const float* w_qkv = (const float*)d_in[1];
  const float* w_out = (const float*)d_in[2];
  const float* b_out = (const float*)d_in[3];
  float* outp = (float*)d_out;

  float* S    = (float*)d_ws;                  // 4096 floats
  float* CTX  = S + BATCH * HEADS * DK;        // 262144 floats
  float* WF   = CTX + BATCH * HEADS * DK * DK; // 1048576 floats
  float* WBIG = WF + (size_t)BATCH * C_IN * DHID;  // 524288 floats

  (void)hipMemsetAsync(d_ws, 0,
                       (size_t)(BATCH * HEADS * DK + BATCH * HEADS * DK * DK) *
                           sizeof(float),
                       stream);

  k1_context<<<dim3(BATCH * HEADS, NPIX / (K1_TS * K1_SUB)), 256, 0, stream>>>(
      img, w_qkv, S, CTX);
  kf1_fold<<<dim3(BATCH * HEADS), 256, 0, stream>>>(w_out, CTX, S, WF);
  kf2_wbig<<<dim3(BATCH, 2), 256, 0, stream>>>(WF, w_qkv, WBIG);
  k3_out<<<dim3(BATCH, NPIX / K3_TB), 256, 0, stream>>>(img, WBIG, b_out, outp);
}